// GCNEncoder_61194694033415
// MI455X (gfx1250) — compile-verified
//
#include <hip/hip_runtime.h>
#include <hip/hip_bf16.h>

typedef float v2f __attribute__((ext_vector_type(2)));
typedef float v8f __attribute__((ext_vector_type(8)));

#define D 128            // feature dim (all layers)
#define WAVE 32

// ---------------------------------------------------------------------------
// Zero-fill kernels
// ---------------------------------------------------------------------------
__global__ void zero_f32(float* __restrict__ p, int n) {
    int i = blockIdx.x * blockDim.x + threadIdx.x;
    if (i < n) p[i] = 0.0f;
}

__global__ void zero_f32x4(float4* __restrict__ p, int n4) {
    int i = blockIdx.x * blockDim.x + threadIdx.x;
    if (i < n4) p[i] = make_float4(0.f, 0.f, 0.f, 0.f);
}

// ---------------------------------------------------------------------------
// Degree accumulation (target index, as in PyG) and dinv = rsqrt(deg + 1)
// ---------------------------------------------------------------------------
__global__ void degree_kernel(const int* __restrict__ eidx, float* __restrict__ deg, int E) {
    int e = blockIdx.x * blockDim.x + threadIdx.x;
    if (e < E) atomicAdd(&deg[eidx[E + e]], 1.0f);   // dst = edge_index[1][e]
}

__global__ void dinv_kernel(float* __restrict__ deg, int n) {
    int i = blockIdx.x * blockDim.x + threadIdx.x;
    if (i < n) deg[i] = rsqrtf(deg[i] + 1.0f);       // in-place: deg -> dinv
}

// ---------------------------------------------------------------------------
// GEMM  H = X @ W   (X:[rows x 128], W:[128 x 128], both row-major, fp32)
// block = 256 threads = 8 waves. Block computes 16 rows x 128 cols.
// Wave w computes the 16x16 tile at columns [16w, 16w+16) via
// v_wmma_f32_16x16x4_f32, K swept in steps of 4 (32 WMMA per wave).
// rows must be a multiple of 16 (100000 = 6250*16) so EXEC is all-ones.
// ---------------------------------------------------------------------------
__global__ __launch_bounds__(256)
void gemm_wmma_f32(const float* __restrict__ X, const float* __restrict__ W,
                   float* __restrict__ H) {
    const int wave = threadIdx.x >> 5;
    const int lane = threadIdx.x & 31;
    const int row0 = blockIdx.x * 16;
    const int col0 = wave * 16;
    const int m    = lane & 15;          // row within A tile / col within B,C tile
    const int koff = (lane >> 4) << 1;   // lanes 0-15 -> K+{0,1}; lanes 16-31 -> K+{2,3}

    const float* xrow = X + (size_t)(row0 + m) * D;

    v8f acc = {0.f, 0.f, 0.f, 0.f, 0.f, 0.f, 0.f, 0.f};

#pragma unroll
    for (int kk = 0; kk < D; kk += 4) {
        v2f a, b;
        a.x = xrow[kk + koff];
        a.y = xrow[kk + koff + 1];
        b.x = W[(size_t)(kk + koff)     * D + col0 + m];
        b.y = W[(size_t)(kk + koff + 1) * D + col0 + m];
        acc = __builtin_amdgcn_wmma_f32_16x16x4_f32(
            /*neg_a=*/false, a, /*neg_b=*/false, b,
            /*c_mod=*/(short)0, acc, /*reuse_a=*/false, /*reuse_b=*/false);
    }

    // C/D layout: VGPR v holds row (v + 8*(lane>=16)), col = lane&15
    const int mbase = (lane >> 4) * 8;
    float* hcol = H + (size_t)row0 * D + col0 + m;
#pragma unroll
    for (int v = 0; v < 8; ++v)
        hcol[(size_t)(v + mbase) * D] = acc[v];
}

// ---------------------------------------------------------------------------
// Edge scatter: one wave per edge. Each lane owns 4 contiguous features.
// agg[dst] += h[src] * dinv[src]*dinv[dst]   (4x global_atomic_add_f32 / lane)
// The 51.2MB agg buffer is L2-resident (192MB L2) -> atomics resolve in L2.
// ---------------------------------------------------------------------------
__global__ __launch_bounds__(256)
void scatter_edges(const float* __restrict__ H, const int* __restrict__ eidx,
                   const float* __restrict__ dinv, float* __restrict__ agg, int E) {
    int wid  = (int)((blockIdx.x * (size_t)blockDim.x + threadIdx.x) >> 5);
    int lane = threadIdx.x & 31;
    if (wid >= E) return;
    int src = eidx[wid];
    int dst = eidx[E + wid];
    float norm = dinv[src] * dinv[dst];

    const float4* hs = (const float4*)(H + (size_t)src * D);
    float4 v = hs[lane];
    float* ad = agg + (size_t)dst * D + lane * 4;
    atomicAdd(ad + 0, v.x * norm);
    atomicAdd(ad + 1, v.y * norm);
    atomicAdd(ad + 2, v.z * norm);
    atomicAdd(ad + 3, v.w * norm);
}

// ---------------------------------------------------------------------------
// Fused self-loop + bias + ReLU:  h = relu(agg + h*dinv^2 + b)
// One thread per float4 (node = i>>5, feature group = i&31).
// ---------------------------------------------------------------------------
__global__ void post_relu(const float4* __restrict__ agg, float4* __restrict__ h,
                          const float* __restrict__ dinv, const float4* __restrict__ b,
                          int n4) {
    int i = blockIdx.x * blockDim.x + threadIdx.x;
    if (i >= n4) return;
    int node = i >> 5;
    int fg   = i & 31;
    float di = dinv[node];
    float d2 = di * di;
    float4 a = agg[i], hh = h[i], bb = b[fg];
    float4 r;
    r.x = fmaxf(a.x + hh.x * d2 + bb.x, 0.f);
    r.y = fmaxf(a.y + hh.y * d2 + bb.y, 0.f);
    r.z = fmaxf(a.z + hh.z * d2 + bb.z, 0.f);
    r.w = fmaxf(a.w + hh.w * d2 + bb.w, 0.f);
    h[i] = r;
}

// ---------------------------------------------------------------------------
// Fused self-loop + bias + LayerNorm. One wave per node; each lane owns 4
// features; mean/var via wave32 shfl_xor tree reduction.
// out (= agg2, in place) := LN(agg2 + h2*dinv^2 + b2) * gamma + beta
// ---------------------------------------------------------------------------
__global__ __launch_bounds__(256)
void post_layernorm(float* __restrict__ out, const float* __restrict__ h2,
                    const float* __restrict__ dinv, const float* __restrict__ b,
                    const float* __restrict__ gamma, const float* __restrict__ beta,
                    int nnodes) {
    int node = (int)((blockIdx.x * (size_t)blockDim.x + threadIdx.x) >> 5);
    int lane = threadIdx.x & 31;
    if (node >= nnodes) return;

    float di = dinv[node];
    float d2 = di * di;

    float4* op = (float4*)(out + (size_t)node * D);
    const float4* hp = (const float4*)(h2 + (size_t)node * D);
    float4 v = op[lane];
    float4 hh = hp[lane];
    float4 bb = ((const float4*)b)[lane];
    v.x += hh.x * d2 + bb.x;
    v.y += hh.y * d2 + bb.y;
    v.z += hh.z * d2 + bb.z;
    v.w += hh.w * d2 + bb.w;

    float s  = v.x + v.y + v.z + v.w;
    float sq = v.x * v.x + v.y * v.y + v.z * v.z + v.w * v.w;
#pragma unroll
    for (int off = 16; off >= 1; off >>= 1) {
        s  += __shfl_xor(s,  off, 32);
        sq += __shfl_xor(sq, off, 32);
    }
    float mean = s * (1.0f / D);
    float var  = sq * (1.0f / D) - mean * mean;
    float inv  = rsqrtf(var + 1e-5f);

    float4 g  = ((const float4*)gamma)[lane];
    float4 be = ((const float4*)beta)[lane];
    float4 r;
    r.x = (v.x - mean) * inv * g.x + be.x;
    r.y = (v.y - mean) * inv * g.y + be.y;
    r.z = (v.z - mean) * inv * g.z + be.z;
    r.w = (v.w - mean) * inv * g.w + be.w;
    op[lane] = r;
}

// ---------------------------------------------------------------------------
// Launcher
// ---------------------------------------------------------------------------
extern "C" void kernel_launch(void* const* d_in, const int* in_sizes, int n_in,
                              void* d_out, int out_size, void* d_ws, size_t ws_size,
                              hipStream_t stream) {
    const float* x     = (const float*)d_in[0];
    const int*   eidx  = (const int*)  d_in[1];   // [2, E] flat: src then dst
    const float* W1    = (const float*)d_in[2];
    const float* b1    = (const float*)d_in[3];
    const float* W2    = (const float*)d_in[4];
    const float* b2    = (const float*)d_in[5];
    const float* gamma = (const float*)d_in[6];
    const float* beta  = (const float*)d_in[7];
    float* out = (float*)d_out;

    const int N = in_sizes[0] / D;        // 100000
    const int E = in_sizes[1] / 2;        // 1600000
    const size_t ND = (size_t)N * D;

    // Workspace layout (floats): dinv | bufA (N*D) | bufB (N*D)
    float* ws   = (float*)d_ws;
    float* dinv = ws;
    float* bufA = ws + (((size_t)N + 255) & ~(size_t)255);
    float* bufB = bufA + ND;

    const int TB = 256;
    const int nd4       = (int)(ND / 4);
    const int grid_nd4  = (nd4 + TB - 1) / TB;
    const int grid_N    = (N + TB - 1) / TB;
    const int grid_E    = (E + TB - 1) / TB;
    const int grid_rows = N / 16;                       // 6250 (N % 16 == 0)
    const int grid_edgewave = (E + (TB / WAVE) - 1) / (TB / WAVE);   // wave per edge
    const int grid_nodewave = (N + (TB / WAVE) - 1) / (TB / WAVE);   // wave per node

    // --- degrees & normalization ---
    zero_f32<<<grid_N, TB, 0, stream>>>(dinv, N);
    degree_kernel<<<grid_E, TB, 0, stream>>>(eidx, dinv, E);
    dinv_kernel<<<grid_N, TB, 0, stream>>>(dinv, N);

    // --- layer 1 ---
    gemm_wmma_f32<<<grid_rows, TB, 0, stream>>>(x, W1, bufA);            // h1 = x @ W1
    zero_f32x4<<<grid_nd4, TB, 0, stream>>>((float4*)bufB, nd4);
    scatter_edges<<<grid_edgewave, TB, 0, stream>>>(bufA, eidx, dinv, bufB, E);
    post_relu<<<grid_nd4, TB, 0, stream>>>((const float4*)bufB, (float4*)bufA,
                                           dinv, (const float4*)b1, nd4); // bufA = relu(...)

    // --- layer 2 ---
    gemm_wmma_f32<<<grid_rows, TB, 0, stream>>>(bufA, W2, bufB);         // h2 = relu @ W2
    zero_f32x4<<<grid_nd4, TB, 0, stream>>>((float4*)out, nd4);
    scatter_edges<<<grid_edgewave, TB, 0, stream>>>(bufB, eidx, dinv, out, E);
    post_layernorm<<<grid_nodewave, TB, 0, stream>>>(out, bufB, dinv, b2,
                                                     gamma, beta, N);
}